// GF_24850680774713
// MI455X (gfx1250) — compile-verified
//
#include <hip/hip_runtime.h>

// ---------------------------------------------------------------------------
// Weighted guided filter for MI455X (gfx1250, wave32).
// Box filters implemented as band-matrix matmuls on the WMMA pipe:
//   Box = M * X * M,  M[i][k] = (|i-k| <= 8)
// Both separable passes are expressed as M * Z so the WMMA A operand is the
// band matrix synthesized from indices (zero loads), B is the data tile.
// HR-resolution kernels are float4 (b128) vectorized: they carry ~96 MB of
// the ~105 MB total HBM traffic (roofline floor ~4.5 us at 23.3 TB/s).
// ---------------------------------------------------------------------------

typedef __attribute__((ext_vector_type(2))) float v2f;
typedef __attribute__((ext_vector_type(8))) float v8f;

#define RR 8
#define LRDIM 256
#define PLANE_ELEMS 65536        // 256*256
#define NPLANE 12                // n*c = 4*3
#define LR_ELEMS 786432          // 12 * 65536
#define HR_ELEMS 12582912        // 12 * 1024*1024
#define EPS_F  1e-8f
#define EPSS_F 1e-12f

__device__ __forceinline__ float cnt_f(int i) {
    int lo = i - RR; if (lo < 0) lo = 0;
    int hi = i + RR; if (hi > LRDIM - 1) hi = LRDIM - 1;
    return (float)(hi - lo + 1);
}
__device__ __forceinline__ float band_f(int i, int k) {
    int d = i - k;
    return (d >= -RR && d <= RR) ? 1.0f : 0.0f;
}

// ------------------------- S = sum(|l_a|) reduction -------------------------
__global__ __launch_bounds__(256) void GF_reduce1(const float4* __restrict__ la4,
                                                  float* __restrict__ partials) {
    __shared__ float sm[256];
    int t = threadIdx.x;
    float s = 0.0f;
    for (int i = blockIdx.x * 256 + t; i < LR_ELEMS / 4; i += 256 * 256) {
        float4 v = la4[i];
        s += fabsf(v.x) + fabsf(v.y) + fabsf(v.z) + fabsf(v.w);
    }
    sm[t] = s; __syncthreads();
    for (int off = 128; off > 0; off >>= 1) {
        if (t < off) sm[t] += sm[t + off];
        __syncthreads();
    }
    if (t == 0) partials[blockIdx.x] = sm[0];
}

__global__ __launch_bounds__(256) void GF_reduce2(const float* __restrict__ partials,
                                                  float* __restrict__ outS) {
    __shared__ float sm[256];
    int t = threadIdx.x;
    sm[t] = partials[t]; __syncthreads();
    for (int off = 128; off > 0; off >>= 1) {
        if (t < off) sm[t] += sm[t + off];
        __syncthreads();
    }
    if (t == 0) outS[0] = sm[0] + EPSS_F * (float)LR_ELEMS;   // S = sum(|la|+eps)
}

// ------------------------- elementwise product planes -----------------------
// B0 layout: quantity q in [0,6): B0[q*LR_ELEMS + i]
// q0=a, q1=a^2 x y, q2=a^2 x (tax pre-scale), q3=a y, q4=a^2 x^2, q5=a x
__global__ __launch_bounds__(256) void GF_products(const float4* __restrict__ lx4,
                                                   const float4* __restrict__ ly4,
                                                   const float4* __restrict__ la4,
                                                   float4* __restrict__ B0) {
    int i = blockIdx.x * 256 + threadIdx.x;
    if (i >= LR_ELEMS / 4) return;
    float4 xa = lx4[i], ya = ly4[i], aa = la4[i];
    float4 q0, q1, q2, q3, q4, q5;
    {
        float a, a2, x, y;
#define DO_COMP(c) \
        a = fabsf(aa.c) + EPSS_F; x = xa.c; y = ya.c; a2 = a * a; \
        q0.c = a; q1.c = a2 * x * y; q2.c = a2 * x; q3.c = a * y; \
        q4.c = a2 * x * x; q5.c = a * x;
        DO_COMP(x) DO_COMP(y) DO_COMP(z) DO_COMP(w)
#undef DO_COMP
    }
    B0[0 * (LR_ELEMS / 4) + i] = q0;
    B0[1 * (LR_ELEMS / 4) + i] = q1;
    B0[2 * (LR_ELEMS / 4) + i] = q2;
    B0[3 * (LR_ELEMS / 4) + i] = q3;
    B0[4 * (LR_ELEMS / 4) + i] = q4;
    B0[5 * (LR_ELEMS / 4) + i] = q5;
}

// ------------------------- WMMA band-matmul box pass ------------------------
// Computes OUT = M * IN per 256x256 plane (one wave per 16x16 tile).
// Pass 1: <false,false>  T1 = M * X            (row-major store)
// Pass 2: <true, true >  tmp = M * T1^T, stored transposed => Box = T1 * M^T,
//                        with the output divided by N = cnt(row)*cnt(col).
template <bool TRANS_IN, bool TRANS_OUT_DIVN>
__global__ __launch_bounds__(32) void GF_boxpass(const float* __restrict__ in,
                                                 float* __restrict__ out) {
    const int lane = threadIdx.x;          // wave32: one wave per block
    const int tile = blockIdx.x;           // 0..255
    const int ti = tile >> 4, tj = tile & 15;
    const float* __restrict__ inp  = in  + (size_t)blockIdx.y * PLANE_ELEMS;
    float* __restrict__       outp = out + (size_t)blockIdx.y * PLANE_ELEMS;

    const int half = lane >> 4;            // 0 or 1
    const int l15  = lane & 15;
    const int arow = 16 * ti + l15;        // A-matrix row held by this lane
    const int bcol = 16 * tj + l15;        // B / output column

    v8f acc = {};
    // Band |i-k|<=8 for output rows [16ti,16ti+15] touches K-blocks ti-1..ti+1
    const int kb0 = (ti > 0)  ? ti - 1 : 0;
    const int kb1 = (ti < 15) ? ti + 1 : 15;
    for (int kb = kb0; kb <= kb1; ++kb) {
#pragma unroll
        for (int c4 = 0; c4 < 4; ++c4) {
            const int k0 = kb * 16 + c4 * 4;
            const int ka = k0 + 2 * half;  // this lane's K pair base (A and B)
            v2f a, b;
            a.x = band_f(arow, ka);
            a.y = band_f(arow, ka + 1);
            if (TRANS_IN) {                // B[k][j] = IN[j][k]: contiguous pair
                b = *(const v2f*)(inp + bcol * LRDIM + ka);   // b64 load
            } else {                       // B[k][j] = IN[k][j]
                b.x = inp[ka * LRDIM + bcol];
                b.y = inp[(ka + 1) * LRDIM + bcol];
            }
            acc = __builtin_amdgcn_wmma_f32_16x16x4_f32(
                false, a, false, b, (short)0, acc, false, false);
        }
    }
#pragma unroll
    for (int v = 0; v < 8; ++v) {
        const int p = 16 * ti + v + 8 * half;  // computed row index of tmp
        const int q = bcol;                    // computed col index of tmp
        float val = acc[v];
        if (TRANS_OUT_DIVN) {
            val *= 1.0f / (cnt_f(q) * cnt_f(p));   // N at final index (q,p)
            outp[q * LRDIM + p] = val;             // store transposed
        } else {
            outp[p * LRDIM + q] = val;
        }
    }
}

// ------------------------- A, b from the six means --------------------------
__global__ __launch_bounds__(256) void GF_ab(const float4* __restrict__ means4,
                                             const float* __restrict__ Sval,
                                             float4* __restrict__ AB4) {
    int t = blockIdx.x * 256 + threadIdx.x;     // quad index
    if (t >= LR_ELEMS / 4) return;
    int base = t * 4;
    int pix = base & (PLANE_ELEMS - 1);
    int i = pix >> 8, j0 = pix & 255;           // quad stays within one row
    float invS = 1.0f / Sval[0];
    float4 m_a    = means4[0 * (LR_ELEMS / 4) + t];
    float4 m_a2xy = means4[1 * (LR_ELEMS / 4) + t];
    float4 m_a2x  = means4[2 * (LR_ELEMS / 4) + t];
    float4 m_ay   = means4[3 * (LR_ELEMS / 4) + t];
    float4 m_a2x2 = means4[4 * (LR_ELEMS / 4) + t];
    float4 m_ax   = means4[5 * (LR_ELEMS / 4) + t];
    float cyi = cnt_f(i);
    float4 Aо, bо;
    {
        float Nf, tax, temp, A, b;
        int c;
#define DO_AB(fld, cc) \
        c = cc; Nf = cyi * cnt_f(j0 + c); tax = m_a2x.fld * invS; \
        temp = fabsf(m_a2x2.fld - Nf * tax * m_ax.fld); \
        A = (m_a2xy.fld - Nf * tax * m_ay.fld) / (temp + EPS_F); \
        b = (m_ay.fld - A * m_ax.fld) / m_a.fld; \
        Aо.fld = A; bо.fld = b;
        DO_AB(x, 0) DO_AB(y, 1) DO_AB(z, 2) DO_AB(w, 3)
#undef DO_AB
    }
    AB4[t] = Aо;                          // planes 0..11  : A
    AB4[(LR_ELEMS / 4) + t] = bо;         // planes 12..23 : b
}

// ------------------ fused bilinear upsample + guided output -----------------
// float4-vectorized along x: quads share y0/y1/wy and never straddle rows.
__global__ __launch_bounds__(256) void GF_final(const float4* __restrict__ hrx4,
                                                const float* __restrict__ Alr,
                                                const float* __restrict__ blr,
                                                float4* __restrict__ out4) {
    int q = blockIdx.x * 256 + threadIdx.x;     // quad index
    if (q >= HR_ELEMS / 4) return;
    int base = q * 4;
    int p   = base >> 20;                  // plane (n*c)
    int pix = base & 1048575;
    int y = pix >> 10, xb = pix & 1023;
    const float sc = 255.0f / 1023.0f;
    float ys = y * sc;
    int y0 = (int)ys;
    int y1 = (y0 + 1 > 255) ? 255 : y0 + 1;
    float wy = ys - (float)y0;
    const float omwy = 1.0f - wy;
    const float* __restrict__ Ap0 = Alr + (size_t)p * PLANE_ELEMS + y0 * 256;
    const float* __restrict__ Ap1 = Alr + (size_t)p * PLANE_ELEMS + y1 * 256;
    const float* __restrict__ bp0 = blr + (size_t)p * PLANE_ELEMS + y0 * 256;
    const float* __restrict__ bp1 = blr + (size_t)p * PLANE_ELEMS + y1 * 256;
    float4 h = hrx4[q];
    float4 o;
    {
        float xs, wx, A, b;
        int x0, x1;
#define DO_PIX(fld, cc) \
        xs = (float)(xb + cc) * sc; x0 = (int)xs; \
        x1 = (x0 + 1 > 255) ? 255 : x0 + 1; wx = xs - (float)x0; \
        A = (Ap0[x0] * omwy + Ap1[x0] * wy) * (1.0f - wx) \
          + (Ap0[x1] * omwy + Ap1[x1] * wy) * wx; \
        b = (bp0[x0] * omwy + bp1[x0] * wy) * (1.0f - wx) \
          + (bp0[x1] * omwy + bp1[x1] * wy) * wx; \
        o.fld = A * h.fld + b;
        DO_PIX(x, 0) DO_PIX(y, 1) DO_PIX(z, 2) DO_PIX(w, 3)
#undef DO_PIX
    }
    out4[q] = o;
}

// ---------------------------------------------------------------------------
extern "C" void kernel_launch(void* const* d_in, const int* in_sizes, int n_in,
                              void* d_out, int out_size, void* d_ws, size_t ws_size,
                              hipStream_t stream) {
    const float* lr_x = (const float*)d_in[0];
    const float* lr_y = (const float*)d_in[1];
    const float* hr_x = (const float*)d_in[2];
    const float* l_a  = (const float*)d_in[3];
    float* out = (float*)d_out;

    float* wsf  = (float*)d_ws;
    float* Sv   = wsf;                         // [0]           : S
    float* part = wsf + 8;                     // [8..263]      : partials
    float* B0   = wsf + 512;                   // 6*LR_ELEMS    : products -> means
    float* B1   = B0 + 6 * (size_t)LR_ELEMS;   // 6*LR_ELEMS    : pass-1 temp
    float* B2   = B1 + 6 * (size_t)LR_ELEMS;   // 2*LR_ELEMS    : A,b -> meanA,meanb
    float* B3   = B2 + 2 * (size_t)LR_ELEMS;   // 2*LR_ELEMS    : pass-1 temp (A,b)

    // 1) deterministic global sum S
    GF_reduce1<<<256, 256, 0, stream>>>((const float4*)l_a, part);
    GF_reduce2<<<1, 256, 0, stream>>>(part, Sv);

    // 2) six product planes (b128 in/out)
    GF_products<<<(LR_ELEMS / 4 + 255) / 256, 256, 0, stream>>>(
        (const float4*)lr_x, (const float4*)lr_y, (const float4*)l_a,
        (float4*)B0);

    // 3) box filter the six quantities with WMMA band-matmuls (72 planes)
    GF_boxpass<false, false><<<dim3(256, 72), 32, 0, stream>>>(B0, B1);
    GF_boxpass<true,  true ><<<dim3(256, 72), 32, 0, stream>>>(B1, B0); // means

    // 4) per-pixel A, b (b128 in/out)
    GF_ab<<<(LR_ELEMS / 4 + 255) / 256, 256, 0, stream>>>(
        (const float4*)B0, Sv, (float4*)B2);

    // 5) box filter A and b (24 planes)
    GF_boxpass<false, false><<<dim3(256, 24), 32, 0, stream>>>(B2, B3);
    GF_boxpass<true,  true ><<<dim3(256, 24), 32, 0, stream>>>(B3, B2); // meanA/meanb

    // 6) fused upsample + output (the only HR-bandwidth kernel, b128 in/out)
    GF_final<<<(HR_ELEMS / 4 + 255) / 256, 256, 0, stream>>>(
        (const float4*)hr_x, B2, B2 + (size_t)LR_ELEMS, (float4*)out);
}